// GATNodeWeigher_21062519620284
// MI455X (gfx1250) — compile-verified
//
#include <hip/hip_runtime.h>
#include <hip/hip_bf16.h>
#include <float.h>

typedef __attribute__((ext_vector_type(2))) float v2f;
typedef __attribute__((ext_vector_type(4))) float v4f;
typedef __attribute__((ext_vector_type(8))) float v8f;

#define D_DIM     128
#define TILE_ROWS 64
#define LSTRIDE   132   // 128 + 4 pad: row r, word k -> bank (4r + k) % 64, conflict-free b64 frags

// ---- order-preserving float <-> uint key (for atomic max on f32) ----
__device__ __forceinline__ unsigned fkey(float f) {
    unsigned u = __float_as_uint(f);
    return (u & 0x80000000u) ? ~u : (u | 0x80000000u);
}
__device__ __forceinline__ float funkey(unsigned k) {
    return (k & 0x80000000u) ? __uint_as_float(k & 0x7fffffffu) : __uint_as_float(~k);
}

__global__ void init_kernel(unsigned* __restrict__ segmax_key, float* __restrict__ segsum, int G) {
    int g = blockIdx.x * blockDim.x + threadIdx.x;
    if (g < G) { segmax_key[g] = fkey(-FLT_MAX); segsum[g] = 0.0f; }
}

// coeff[d]     = sum_{k<256} W_fc[k,d] * W_attn[k]   (d < 128)
// coeff[128+f] = W_attn[256+f]                        (f < 128)
__global__ void coeff_kernel(const float* __restrict__ W_fc, const float* __restrict__ W_attn,
                             float* __restrict__ coeff) {
    int t = threadIdx.x;           // 256 threads
    if (t < 128) {
        float acc = 0.0f;
        for (int k = 0; k < 256; ++k) acc = fmaf(W_fc[k * 128 + t], W_attn[k], acc);
        coeff[t] = acc;
    } else {
        coeff[t] = W_attn[t + 128]; // coeff[128+f] = W_attn[256+f]
    }
}

// One block = 128 threads = 4 waves; each wave computes 16 rows' scores via
// V_WMMA_F32_16X16X4_F32 accumulation over K=256 (weights then feats).
// Two independent accumulators (even/odd K chunks) break the WMMA RAW chain.
__global__ __launch_bounds__(128) void score_kernel(
    const float* __restrict__ wx, const float* __restrict__ fx,
    const float* __restrict__ coeff, const int* __restrict__ seg,
    float* __restrict__ score, unsigned* __restrict__ segmax_key)
{
    __shared__ float sc[256];
    __shared__ float tile[TILE_ROWS * LSTRIDE];

    const int t    = threadIdx.x;
    const int lane = t & 31;
    const int wave = t >> 5;
    const int row  = lane & 15;   // A-frag: lane m and lane m+16 both hold row m
    const int half = lane >> 4;   // lanes 0-15 -> K{0,1}; lanes 16-31 -> K{2,3}
    const long long base = (long long)blockIdx.x * TILE_ROWS;

    for (int i = t; i < 256; i += 128) sc[i] = coeff[i];

    v8f acc0 = (v8f)0.0f;
    v8f acc1 = (v8f)0.0f;

    #pragma unroll
    for (int ph = 0; ph < 2; ++ph) {
        const float* src = (ph == 0 ? wx : fx) + base * D_DIM;
        __syncthreads();   // sc ready (ph 0) / previous phase compute done (ph 1)
        // Stage 64x128 f32 tile (contiguous 32KB block) coalesced: b128 loads.
        for (int j = t; j < (TILE_ROWS * D_DIM / 4); j += 128) {
            v4f v = ((const v4f*)src)[j];
            int r = j >> 5;            // 32 float4 per row
            int c = (j & 31) << 2;
            *(v4f*)&tile[r * LSTRIDE + c] = v;
        }
        __syncthreads();

        const float* arow = &tile[(wave * 16 + row) * LSTRIDE + 2 * half];
        const float* brow = &sc[ph * 128 + 2 * half];
        #pragma unroll
        for (int k0 = 0; k0 < 128; k0 += 8) {
            // A (16x4): lane m -> A[m, k0+2h .. k0+2h+1]     (ISA 16x4 f32 layout)
            v2f a0 = *(const v2f*)(arow + k0);
            v2f b0 = *(const v2f*)(brow + k0);
            v2f a1 = *(const v2f*)(arow + k0 + 4);
            v2f b1 = *(const v2f*)(brow + k0 + 4);
            acc0 = __builtin_amdgcn_wmma_f32_16x16x4_f32(
                       false, a0, false, b0, (short)0, acc0, false, false);
            acc1 = __builtin_amdgcn_wmma_f32_16x16x4_f32(
                       false, a1, false, b1, (short)0, acc1, false, false);
        }
    }

    // C/D layout: c[r] lanes 0-15 hold M=r, lanes 16-31 hold M=8+r (all N columns equal).
    if (row == 0) {
        long long mbase = base + wave * 16 + half * 8;
        int curseg = -1; float curmax = -FLT_MAX;
        #pragma unroll
        for (int r = 0; r < 8; ++r) {
            float s = acc0[r] + acc1[r];
            s = (s >= 0.0f) ? s : 0.1f * s;          // LeakyReLU(0.1)
            long long node = mbase + r;
            score[node] = s;
            int sg = seg[node];
            if (sg != curseg) {
                if (curseg >= 0) atomicMax(&segmax_key[curseg], fkey(curmax));
                curseg = sg; curmax = s;
            } else {
                curmax = fmaxf(curmax, s);
            }
        }
        atomicMax(&segmax_key[curseg], fkey(curmax));
    }
}

// Defensive scalar path for N % 64 leftover rows (unused for N = 1e6).
__global__ void score_tail_kernel(const float* __restrict__ wx, const float* __restrict__ fx,
                                  const float* __restrict__ coeff, const int* __restrict__ seg,
                                  float* __restrict__ score, unsigned* __restrict__ segmax_key,
                                  int start, int N) {
    int i = start + blockIdx.x * blockDim.x + threadIdx.x;
    if (i >= N) return;
    float a = 0.0f;
    for (int k = 0; k < 128; ++k) a = fmaf(wx[(long long)i * 128 + k], coeff[k], a);
    for (int k = 0; k < 128; ++k) a = fmaf(fx[(long long)i * 128 + k], coeff[128 + k], a);
    float s = (a >= 0.0f) ? a : 0.1f * a;
    score[i] = s;
    atomicMax(&segmax_key[seg[i]], fkey(s));
}

__global__ void expsum_kernel(const float* __restrict__ score, const int* __restrict__ seg,
                              const unsigned* __restrict__ segmax_key,
                              float* __restrict__ segsum, int N) {
    long long i0 = (long long)(blockIdx.x * blockDim.x + threadIdx.x) * 8;
    if (i0 >= N) return;
    int lim = (int)(((long long)N - i0) < 8 ? ((long long)N - i0) : 8);
    int curseg = -1; float acc = 0.0f, curmax = 0.0f;
    for (int r = 0; r < lim; ++r) {
        long long i = i0 + r;
        int sg = seg[i];
        if (sg != curseg) {
            if (curseg >= 0) atomicAdd(&segsum[curseg], acc);
            curseg = sg; acc = 0.0f; curmax = funkey(segmax_key[sg]);
        }
        acc += expf(score[i] - curmax);
    }
    atomicAdd(&segsum[curseg], acc);
}

__global__ void norm_kernel(float* __restrict__ out, const int* __restrict__ seg,
                            const unsigned* __restrict__ segmax_key,
                            const float* __restrict__ segsum, float scale, int N) {
    int i = blockIdx.x * blockDim.x + threadIdx.x;
    if (i >= N) return;
    int sg = seg[i];
    float m = funkey(segmax_key[sg]);
    out[i] = expf(out[i] - m) / segsum[sg] * scale;   // in place: out held the score
}

extern "C" void kernel_launch(void* const* d_in, const int* in_sizes, int n_in,
                              void* d_out, int out_size, void* d_ws, size_t ws_size,
                              hipStream_t stream) {
    const float* wx     = (const float*)d_in[0];
    const float* fx     = (const float*)d_in[1];
    const float* W_fc   = (const float*)d_in[2];
    const float* W_attn = (const float*)d_in[3];
    const int*   seg    = (const int*)d_in[4];

    const int N = in_sizes[0] / D_DIM;   // 1,000,000
    const int G = 512;                   // reference num_graphs (fixed in module)

    float*    score      = (float*)d_out;                                    // N scores, normalized in place
    float*    coeff      = (float*)d_ws;                                     // 256 f32
    unsigned* segmax_key = (unsigned*)((char*)d_ws + 256 * sizeof(float));   // G u32
    float*    segsum     = (float*)((char*)d_ws + (256 + 512) * sizeof(float)); // G f32

    init_kernel<<<(G + 255) / 256, 256, 0, stream>>>(segmax_key, segsum, G);
    coeff_kernel<<<1, 256, 0, stream>>>(W_fc, W_attn, coeff);

    int tiles = N / TILE_ROWS;
    if (tiles > 0)
        score_kernel<<<tiles, 128, 0, stream>>>(wx, fx, coeff, seg, score, segmax_key);
    int rem_start = tiles * TILE_ROWS;
    if (rem_start < N)
        score_tail_kernel<<<(N - rem_start + 255) / 256, 256, 0, stream>>>(
            wx, fx, coeff, seg, score, segmax_key, rem_start, N);

    int nthr = (N + 7) / 8;
    expsum_kernel<<<(nthr + 255) / 256, 256, 0, stream>>>(score, seg, segmax_key, segsum, N);

    float scale = (float)N / (float)G;   // mean node count per graph, exactly
    norm_kernel<<<(N + 255) / 256, 256, 0, stream>>>(score, seg, segmax_key, segsum, scale, N);
}